// GVQ_VAE_codebook_loss_41729902248240
// MI455X (gfx1250) — compile-verified
//
#include <hip/hip_runtime.h>
#include <math.h>

// Problem constants (from reference)
#define S_CODES 2048
#define C_DIM   512
#define P_DIM   1024
#define N_BATCH 32
#define TOPK    16          // e^-16 ~ 1e-7 relative: beyond this, weights vanish in fp32
#define CHUNK_K   128       // codebook k-chunk (floats per row)
#define CB_STRIDE 132       // 128 + 4 pad words -> stride mod 64 == 4 (bank-conflict-free A reads)
#define NCHUNK    (S_CODES / 16 * (C_DIM / CHUNK_K))   // 512 chunks total

typedef float v2f __attribute__((ext_vector_type(2)));
typedef float v8f __attribute__((ext_vector_type(8)));

// ---------------------------------------------------------------------------
// Kernel 1: c2[s] = sum_c codebook[s,c]^2   (one wave per codebook row)
// ---------------------------------------------------------------------------
__global__ __launch_bounds__(32) void k_c2(const float* __restrict__ cb,
                                           float* __restrict__ c2) {
  const int s = blockIdx.x;
  const int l = threadIdx.x;
  float v = 0.f;
#pragma unroll
  for (int j = 0; j < C_DIM / 32; ++j) {
    float t = cb[(size_t)s * C_DIM + l + j * 32];
    v += t * t;
  }
#pragma unroll
  for (int off = 16; off >= 1; off >>= 1) v += __shfl_xor(v, off, 32);
  if (l == 0) c2[s] = v;
}

// ---------------------------------------------------------------------------
// Kernel 2: fused distance-GEMM (WMMA f32 16x16x4) + argmin + weighted top-K
//           + codebook gather (straight-through output) + partial losses.
// Codebook tiles are streamed into LDS with double-buffered
// GLOBAL_LOAD_ASYNC_TO_LDS_B128 (ASYNCcnt), overlapping DMA with WMMA.
// One wave per (n, 16-column p-tile). Grid = 32 * 64 = 2048 waves.
// ---------------------------------------------------------------------------
__global__ __launch_bounds__(32) void k_main(const float* __restrict__ x,
                                             const float* __restrict__ cb,
                                             const float* __restrict__ c2g,
                                             float* __restrict__ outBuf,
                                             float* __restrict__ partials) {
  __shared__ float xL[C_DIM * 16];            // -2*x tile, [c][16 cols]   32 KB
  __shared__ float cbL[2][16 * CB_STRIDE];    // double-buffered cb chunks 16.9 KB
  __shared__ float c2L[S_CODES];              // c2 broadcast               8 KB
  __shared__ float topL[16 * 2 * TOPK];       // per-column 2 sorted lists
  __shared__ float minvL[32];
  __shared__ int   miniL[32];
  __shared__ int   idxL[16];
  __shared__ float redL[32];

  const int n     = blockIdx.x >> 6;          // 64 p-tiles per batch image
  const int p0    = (blockIdx.x & 63) * 16;
  const int l     = threadIdx.x;
  const int col   = l & 15;                   // WMMA N index / cb row in chunk
  const int khalf = l >> 4;                   // WMMA K half selector

  // Per-lane base LDS byte offsets of the two chunk buffers (low 32 bits of a
  // generic LDS pointer are the wave-relative LDS byte address).
  const unsigned ldsBase0 =
      (unsigned)(size_t)&cbL[0][col * CB_STRIDE + khalf * (CHUNK_K / 2)];
  const unsigned ldsBase1 =
      (unsigned)(size_t)&cbL[1][col * CB_STRIDE + khalf * (CHUNK_K / 2)];

  // Issue one 16-row x CHUNK_K codebook chunk as 16 async b128 DMAs per lane.
  // chunk t: s0 = (t>>2)*16, kc = (t&3)*CHUNK_K. Lane covers row (l&15),
  // 64 floats selected by khalf -> 256 contiguous bytes on both sides.
  auto issue_chunk = [&](int t, int buf) {
    const int s0 = (t >> 2) * 16;
    const int kc = (t & 3) * CHUNK_K;
    unsigned lds = buf ? ldsBase1 : ldsBase0;
    unsigned long long g =
        (unsigned long long)(cb + (size_t)(s0 + col) * C_DIM + kc + khalf * (CHUNK_K / 2));
#pragma unroll
    for (int q = 0; q < 16; ++q) {
      asm volatile("global_load_async_to_lds_b128 %0, %1, off"
                   :: "v"(lds), "v"(g) : "memory");
      lds += 16;
      g += 16;
    }
  };

  issue_chunk(0, 0);                          // prefetch first chunk ASAP

  // ---- stage x tile scaled by -2 (folds the -2 of d = x2+c2-2xc into B) ----
  for (int c = l; c < C_DIM; c += 32) {
    const float* src = x + ((size_t)n * C_DIM + c) * P_DIM + p0;
#pragma unroll
    for (int q = 0; q < 4; ++q) {
      float4 v = *(const float4*)(src + q * 4);
      v.x *= -2.f; v.y *= -2.f; v.z *= -2.f; v.w *= -2.f;
      *(float4*)&xL[c * 16 + q * 4] = v;
    }
  }
  for (int i = l; i < S_CODES; i += 32) c2L[i] = c2g[i];
  __syncthreads();

  // per-column x^2: xL holds -2x, so sum(xL^2) = 4*sum(x^2)
  float x2v = 0.f;
  for (int c = 0; c < C_DIM; ++c) { float t = xL[c * 16 + col]; x2v += t * t; }

  float tk[TOPK];                             // per-lane sorted smallest-e list
#pragma unroll
  for (int j = 0; j < TOPK; ++j) tk[j] = 3.0e38f;
  float minv = 3.0e38f;
  int   mini = 0;

  // ---- main loop: e[s,col] = c2[s] - 2*codebook[s,:]·x[:,col] via WMMA ----
  v8f acc;
  for (int t = 0; t < NCHUNK; ++t) {
    const int buf = t & 1;
    const int s0  = (t >> 2) * 16;
    const int kc  = (t & 3) * CHUNK_K;

    if ((t & 3) == 0) {                       // new s-tile: C init with c2[s]
#pragma unroll
      for (int r = 0; r < 8; ++r) acc[r] = c2L[s0 + khalf * 8 + r];
    }

    if (t + 1 < NCHUNK) {
      issue_chunk(t + 1, buf ^ 1);            // DMA next chunk into other buffer
      // async loads complete in order: <=16 outstanding => chunk t fully landed
      asm volatile("s_wait_asynccnt 0x10" ::: "memory");
    } else {
      asm volatile("s_wait_asynccnt 0x0" ::: "memory");
    }

    const float* arow = &cbL[buf][col * CB_STRIDE];
#pragma unroll 8
    for (int k = 0; k < CHUNK_K; k += 4) {
      const int kk = k + 2 * khalf;           // ISA A/B layout: K pair by lane half
      v2f a, b;
      a.x = arow[kk];                         // A[M=col][K=kk], A[M=col][K=kk+1]
      a.y = arow[kk + 1];
      b.x = xL[(kc + kk) * 16 + col];         // B[K=kk][N=col]   (= -2x)
      b.y = xL[(kc + kk + 1) * 16 + col];     // B[K=kk+1][N=col]
      acc = __builtin_amdgcn_wmma_f32_16x16x4_f32(
          false, a, false, b, (short)0, acc, false, false);
    }

    if ((t & 3) == 3) {                       // s-tile done: argmin + top-K
      const int sbase = s0 + khalf * 8;
#pragma unroll
      for (int r = 0; r < 8; ++r) {
        float v = acc[r];
        if (v < minv) { minv = v; mini = sbase + r; }   // strict '<': first hit
        if (v < tk[TOPK - 1]) {                         // rare after warm-up
#pragma unroll
          for (int j = TOPK - 1; j >= 1; --j)
            tk[j] = (v < tk[j - 1]) ? tk[j - 1] : ((v < tk[j]) ? v : tk[j]);
          tk[0] = (v < tk[0]) ? v : tk[0];
        }
      }
    }
  }

  // ---- per-column combine of the two lane halves ----
#pragma unroll
  for (int j = 0; j < TOPK; ++j) topL[col * (2 * TOPK) + khalf * TOPK + j] = tk[j];
  minvL[l] = minv; miniL[l] = mini;
  __syncthreads();

  float colLoss = 0.f;
  if (l < 16) {
    float mva = minvL[l], mvb = minvL[l + 16];
    int   mia = miniL[l], mib = miniL[l + 16];
    int best = (mvb < mva || (mvb == mva && mib < mia)) ? mib : mia;
    idxL[l] = best;
    // merge two sorted TOPK lists; weighted sum of smallest TOPK e-values.
    const float* A = &topL[l * (2 * TOPK)];
    const float* B = A + TOPK;
    int i = 0, j = 0;
    float ws = 0.f;
    for (int t = 0; t < TOPK; ++t) {
      float av = A[i], bv = B[j];
      float m;
      if (av <= bv) { m = av; ++i; } else { m = bv; ++j; }
      ws += __expf(-(float)t) * m;
    }
    // sum_s e^-s over s=0..2047  ==  1/(1-e^-1) (tail underflows fp32)
    const float WTOT = 1.5819767068693265f;
    colLoss = (0.25f * x2v) * WTOT + ws;      // undo the (-2)^2 on x^2
  }
  redL[l] = colLoss;                          // lanes >=16 contribute 0
  __syncthreads();
  if (l == 0) {
    float s = 0.f;
    for (int t = 0; t < 16; ++t) s += redL[t];
    partials[(size_t)blockIdx.x * 2] = s;     // codebook-loss partial
  }
  if (l < 16) outBuf[2 + n * P_DIM + p0 + l] = (float)idxL[l];   // indices out
  __syncthreads();

  // ---- straight-through output xq + commitment loss partial ----
  float* outq = outBuf + 2 + (size_t)N_BATCH * P_DIM;
  const int bidx = idxL[col];
  float csum = 0.f;
  for (int cc = 0; cc < 256; ++cc) {
    int c = cc * 2 + khalf;                   // halves interleave c -> coalesced stores
    float qv = cb[(size_t)bidx * C_DIM + c];  // L2-hot gather
    float xv = -0.5f * xL[c * 16 + col];      // xL holds -2x
    float df = xv - qv;
    csum += df * df;
    outq[((size_t)n * C_DIM + c) * P_DIM + p0 + col] = qv;
  }
  redL[l] = csum;
  __syncthreads();
  if (l == 0) {
    float s = 0.f;
    for (int t = 0; t < 32; ++t) s += redL[t];
    partials[(size_t)blockIdx.x * 2 + 1] = s; // commitment partial
  }
}

// ---------------------------------------------------------------------------
// Kernel 3: deterministic final reduction of the 2048 partial pairs.
// ---------------------------------------------------------------------------
__global__ __launch_bounds__(256) void k_reduce(const float* __restrict__ partials,
                                                float* __restrict__ outBuf) {
  __shared__ float s0[256], s1[256];
  const int t = threadIdx.x;
  float a = 0.f, b = 0.f;
  for (int i = t; i < N_BATCH * 64; i += 256) {
    a += partials[2 * i];
    b += partials[2 * i + 1];
  }
  s0[t] = a; s1[t] = b;
  __syncthreads();
  for (int off = 128; off >= 1; off >>= 1) {
    if (t < off) { s0[t] += s0[t + off]; s1[t] += s1[t + off]; }
    __syncthreads();
  }
  if (t == 0) {
    outBuf[0] = s0[0] / (float)((size_t)N_BATCH * S_CODES * P_DIM);  // /67108864
    outBuf[1] = s1[0] / (float)(N_BATCH * C_DIM * P_DIM);            // /16777216
  }
}

// ---------------------------------------------------------------------------
extern "C" void kernel_launch(void* const* d_in, const int* in_sizes, int n_in,
                              void* d_out, int out_size, void* d_ws, size_t ws_size,
                              hipStream_t stream) {
  const float* x  = (const float*)d_in[0];   // [32,512,32,32]
  const float* cb = (const float*)d_in[1];   // [2048,512]
  float* out      = (float*)d_out;           // [loss_cb, loss_commit, idx(32768), xq(16777216)]
  float* c2       = (float*)d_ws;            // 2048 floats
  float* partials = c2 + S_CODES;            // 2048 * 2 floats

  k_c2    <<<dim3(S_CODES),      dim3(32),  0, stream>>>(cb, c2);
  k_main  <<<dim3(N_BATCH * 64), dim3(32),  0, stream>>>(x, cb, c2, out, partials);
  k_reduce<<<dim3(1),            dim3(256), 0, stream>>>(partials, out);
}